// DependencyGNN_45500883534516
// MI455X (gfx1250) — compile-verified
//
#include <hip/hip_runtime.h>
#include <math.h>

#define N_NODES 10000
#define N_EDGES 320000
#define IN_F    256
#define HID     128
#define HEADS   4
#define ZDIM    (HEADS * HID)   // 512

typedef __attribute__((ext_vector_type(2))) float v2f;
typedef __attribute__((ext_vector_type(8))) float v8f;

// ---------------------------------------------------------------- utilities
__global__ void gnn_fill_f32(float* __restrict__ p, float v, int n) {
    int i = blockIdx.x * blockDim.x + threadIdx.x;
    if (i < n) p[i] = v;
}

__global__ void gnn_deg_count(const int* __restrict__ src, const int* __restrict__ dst,
                              float* __restrict__ dout, float* __restrict__ din, int E) {
    int e = blockIdx.x * blockDim.x + threadIdx.x;
    if (e >= E) return;
    atomicAdd(&dout[src[e]], 1.0f);
    atomicAdd(&din[dst[e]],  1.0f);
}

__global__ void gnn_deg_finalize(float* __restrict__ d, int n) {
    int i = blockIdx.x * blockDim.x + threadIdx.x;
    if (i >= n) return;
    float v = fmaxf(d[i], 1.0f);
    d[i] = 1.0f / sqrtf(v);     // deg^-1/2 after clip(., 1.0)
}

// ------------------------------------------------------ fp32 WMMA GEMM core
// C[M,Nout] = (A .* rowScale) @ W ; A row-major [M,K], W row-major [K,Nout].
// One wave per 16x16 tile, K consumed 4 at a time via V_WMMA_F32_16X16X4_F32.
// A-frag layout (ISA 7.12.2, 32-bit A 16x4): lanes 0-15 -> K,K+1 ; 16-31 -> K+2,K+3.
// C/D layout: VGPR v, lanes 0-15 -> row v ; lanes 16-31 -> row v+8.
__global__ void gnn_gemm_wmma_f32(const float* __restrict__ A,
                                  const float* __restrict__ W,
                                  const float* __restrict__ rowScale,
                                  float* __restrict__ C,
                                  int M, int K, int Nout) {
    int wave = (int)((blockIdx.x * blockDim.x + threadIdx.x) >> 5);
    int lane = threadIdx.x & 31;
    int tilesN = Nout >> 4;
    int tilesM = (M + 15) >> 4;
    if (wave >= tilesM * tilesN) return;          // wave-uniform exit (EXEC stays all-1)
    int tm = wave / tilesN;
    int tn = wave - tm * tilesN;

    int l15  = lane & 15;
    int kh   = (lane >> 4) << 1;                  // 0 for lanes 0-15, 2 for 16-31
    int aRow = tm * 16 + l15;
    const float* Ap = A + (size_t)aRow * K;
    float sc = rowScale ? rowScale[aRow] : 1.0f;
    const float* Wp = W + tn * 16 + l15;

    v8f acc = {};
    for (int k = 0; k < K; k += 4) {
        v2f a, b;
        a.x = Ap[k + kh]     * sc;
        a.y = Ap[k + kh + 1] * sc;
        b.x = Wp[(size_t)(k + kh)     * Nout];
        b.y = Wp[(size_t)(k + kh + 1) * Nout];
        acc = __builtin_amdgcn_wmma_f32_16x16x4_f32(
            /*neg_a=*/false, a, /*neg_b=*/false, b,
            /*c_mod=*/(short)0, acc, /*reuse_a=*/false, /*reuse_b=*/false);
    }
    int rbase = tm * 16 + ((lane >> 4) << 3);
    float* Cp = C + (size_t)rbase * Nout + tn * 16 + l15;
#pragma unroll
    for (int v = 0; v < 8; ++v) Cp[(size_t)v * Nout] = acc[v];
}

// ------------------------------------------------------------ edge kernels
// One wave per edge, 128 features, 4 per lane; fp32 atomics resolve in L2.
__global__ void gnn_edge_agg128(const float* __restrict__ h, const int* __restrict__ src,
                                const int* __restrict__ dst, float* __restrict__ agg, int E) {
    int wave = (int)((blockIdx.x * blockDim.x + threadIdx.x) >> 5);
    int lane = threadIdx.x & 31;
    if (wave >= E) return;
    const float* hp = h + (size_t)src[wave] * HID;
    float* ap = agg + (size_t)dst[wave] * HID;
#pragma unroll
    for (int i = 0; i < 4; ++i) {
        int f = lane + 32 * i;
        atomicAdd(&ap[f], hp[f]);
    }
}

__global__ void gnn_post_norm_relu(const float* __restrict__ agg, const float* __restrict__ dins,
                                   const float* __restrict__ b, float* __restrict__ out) {
    int i = blockIdx.x * blockDim.x + threadIdx.x;
    if (i >= N_NODES * HID) return;
    int n = i >> 7, f = i & (HID - 1);
    float v = agg[i] * dins[n] + b[f];
    out[i] = v > 0.0f ? v : 0.0f;
}

// ------------------------------------------------------------- GAT kernels
__global__ void gnn_attn_proj(const float* __restrict__ z, const float* __restrict__ al,
                              const float* __restrict__ ar, float* __restrict__ el,
                              float* __restrict__ er) {
    int i = blockIdx.x * blockDim.x + threadIdx.x;   // n*HEADS + h
    if (i >= N_NODES * HEADS) return;
    int n = i >> 2, hh = i & 3;
    const float* zp = z + (size_t)n * ZDIM + hh * HID;
    const float* lp = al + hh * HID;
    const float* rp = ar + hh * HID;
    float sl = 0.0f, sr = 0.0f;
#pragma unroll 8
    for (int f = 0; f < HID; ++f) { float zv = zp[f]; sl += zv * lp[f]; sr += zv * rp[f]; }
    el[i] = sl; er[i] = sr;
}

__device__ __forceinline__ void atomicMaxF(float* addr, float val) {
    if (val >= 0.0f) atomicMax((int*)addr, __float_as_int(val));
    else             atomicMin((unsigned int*)addr, __float_as_uint(val));
}

__global__ void gnn_edge_scores(const float* __restrict__ el, const float* __restrict__ er,
                                const int* __restrict__ src, const int* __restrict__ dst,
                                float* __restrict__ eedge, float* __restrict__ mmax, int E) {
    int i = blockIdx.x * blockDim.x + threadIdx.x;   // e*HEADS + h
    if (i >= E * HEADS) return;
    int e = i >> 2, hh = i & 3;
    float v = el[src[e] * HEADS + hh] + er[dst[e] * HEADS + hh];
    v = v > 0.0f ? v : 0.2f * v;                     // leaky_relu(0.2)
    eedge[i] = v;
    atomicMaxF(&mmax[dst[e] * HEADS + hh], v);
}

__global__ void gnn_edge_exp(const int* __restrict__ dst, const float* __restrict__ mmax,
                             float* __restrict__ eedge, float* __restrict__ den, int E) {
    int i = blockIdx.x * blockDim.x + threadIdx.x;
    if (i >= E * HEADS) return;
    int e = i >> 2, hh = i & 3;
    float ex = expf(eedge[i] - mmax[dst[e] * HEADS + hh]);
    eedge[i] = ex;
    atomicAdd(&den[dst[e] * HEADS + hh], ex);
}

__global__ void gnn_edge_alpha(const int* __restrict__ dst, const float* __restrict__ den,
                               float* __restrict__ eedge, int E) {
    int i = blockIdx.x * blockDim.x + threadIdx.x;
    if (i >= E * HEADS) return;
    int e = i >> 2, hh = i & 3;
    eedge[i] = eedge[i] / den[dst[e] * HEADS + hh];
}

// One wave per (edge, head); alpha read once per wave, 4 feats per lane.
__global__ void gnn_gat_agg(const float* __restrict__ z, const float* __restrict__ alpha,
                            const int* __restrict__ src, const int* __restrict__ dst,
                            float* __restrict__ oagg, int E) {
    int wave = (int)((blockIdx.x * blockDim.x + threadIdx.x) >> 5);
    int lane = threadIdx.x & 31;
    if (wave >= E * HEADS) return;
    int e = wave >> 2, hh = wave & 3;
    float a = alpha[e * HEADS + hh];
    const float* zp = z + (size_t)src[e] * ZDIM + hh * HID;
    float* op = oagg + (size_t)dst[e] * ZDIM + hh * HID;
#pragma unroll
    for (int i = 0; i < 4; ++i) {
        int f = lane + 32 * i;
        atomicAdd(&op[f], a * zp[f]);
    }
}

__global__ void gnn_head_mean(const float* __restrict__ oagg, const float* __restrict__ bg,
                              float* __restrict__ hm) {
    int i = blockIdx.x * blockDim.x + threadIdx.x;   // n*HID + f
    if (i >= N_NODES * HID) return;
    int n = i >> 7, f = i & (HID - 1);
    float s = 0.0f;
#pragma unroll
    for (int hh = 0; hh < HEADS; ++hh)
        s += oagg[(size_t)n * ZDIM + hh * HID + f] + bg[hh * HID + f];
    hm[i] = 0.25f * s;
}

// ----------------------------------------------------------- final layer
__global__ void gnn_conv3_dot(const float* __restrict__ hm, const float* __restrict__ douts,
                              const float* __restrict__ W3, float* __restrict__ sN) {
    int n = blockIdx.x * blockDim.x + threadIdx.x;
    if (n >= N_NODES) return;
    const float* hp = hm + (size_t)n * HID;
    float s = 0.0f;
#pragma unroll 8
    for (int f = 0; f < HID; ++f) s += hp[f] * W3[f];
    sN[n] = s * douts[n];
}

__global__ void gnn_edge_agg1(const float* __restrict__ sN, const int* __restrict__ src,
                              const int* __restrict__ dst, float* __restrict__ sAgg, int E) {
    int e = blockIdx.x * blockDim.x + threadIdx.x;
    if (e >= E) return;
    atomicAdd(&sAgg[dst[e]], sN[src[e]]);
}

__global__ void gnn_final_sigmoid(const float* __restrict__ sAgg, const float* __restrict__ dins,
                                  const float* __restrict__ b3, float* __restrict__ out) {
    int n = blockIdx.x * blockDim.x + threadIdx.x;
    if (n >= N_NODES) return;
    float v = sAgg[n] * dins[n] + b3[0];
    out[n] = 1.0f / (1.0f + expf(-v));
}

// -------------------------------------------------------------- launcher
extern "C" void kernel_launch(void* const* d_in, const int* in_sizes, int n_in,
                              void* d_out, int out_size, void* d_ws, size_t ws_size,
                              hipStream_t stream) {
    const float* feat = (const float*)d_in[0];
    const int*   src  = (const int*)  d_in[1];
    const int*   dst  = (const int*)  d_in[2];
    const float* W1   = (const float*)d_in[3];
    const float* b1   = (const float*)d_in[4];
    const float* W2   = (const float*)d_in[5];
    const float* b2   = (const float*)d_in[6];
    const float* W3   = (const float*)d_in[7];
    const float* b3   = (const float*)d_in[8];
    const float* Wg   = (const float*)d_in[9];
    const float* al   = (const float*)d_in[10];
    const float* ar   = (const float*)d_in[11];
    const float* bg   = (const float*)d_in[12];
    float* out = (float*)d_out;
    (void)in_sizes; (void)n_in; (void)out_size; (void)ws_size;

    // workspace carve-up (floats)
    float* ws = (float*)d_ws;
    size_t off = 0;
    float* douts = ws + off; off += N_NODES;
    float* dins  = ws + off; off += N_NODES;
    float* t0    = ws + off; off += (size_t)N_NODES * HID;
    float* t1    = ws + off; off += (size_t)N_NODES * HID;
    float* t2    = ws + off; off += (size_t)N_NODES * HID;
    float* zbuf  = ws + off; off += (size_t)N_NODES * ZDIM;
    float* oagg  = ws + off; off += (size_t)N_NODES * ZDIM;
    float* el    = ws + off; off += N_NODES * HEADS;
    float* er    = ws + off; off += N_NODES * HEADS;
    float* mmax  = ws + off; off += N_NODES * HEADS;
    float* den   = ws + off; off += N_NODES * HEADS;
    float* eedge = ws + off; off += (size_t)N_EDGES * HEADS;
    float* sN    = ws + off; off += N_NODES;
    float* sAgg  = ws + off; off += N_NODES;

    const int T = 256;
    auto nb = [](int n, int t) { return (n + t - 1) / t; };

    // degrees
    gnn_fill_f32<<<nb(N_NODES, T), T, 0, stream>>>(douts, 0.0f, N_NODES);
    gnn_fill_f32<<<nb(N_NODES, T), T, 0, stream>>>(dins,  0.0f, N_NODES);
    gnn_deg_count<<<nb(N_EDGES, T), T, 0, stream>>>(src, dst, douts, dins, N_EDGES);
    gnn_deg_finalize<<<nb(N_NODES, T), T, 0, stream>>>(douts, N_NODES);
    gnn_deg_finalize<<<nb(N_NODES, T), T, 0, stream>>>(dins,  N_NODES);

    // GraphConv 1: t0 = (feat .* douts) @ W1 ; agg -> t1 ; relu -> t0
    {
        int waves = (N_NODES / 16) * (HID / 16);          // 625*8 = 5000
        gnn_gemm_wmma_f32<<<nb(waves * 32, T), T, 0, stream>>>(feat, W1, douts, t0,
                                                               N_NODES, IN_F, HID);
        gnn_fill_f32<<<nb(N_NODES * HID, T), T, 0, stream>>>(t1, 0.0f, N_NODES * HID);
        gnn_edge_agg128<<<nb(N_EDGES * 32, T), T, 0, stream>>>(t0, src, dst, t1, N_EDGES);
        gnn_post_norm_relu<<<nb(N_NODES * HID, T), T, 0, stream>>>(t1, dins, b1, t0);
    }
    // GraphConv 2: t1 = (t0 .* douts) @ W2 ; agg -> t2 ; relu -> t0
    {
        int waves = (N_NODES / 16) * (HID / 16);
        gnn_gemm_wmma_f32<<<nb(waves * 32, T), T, 0, stream>>>(t0, W2, douts, t1,
                                                               N_NODES, HID, HID);
        gnn_fill_f32<<<nb(N_NODES * HID, T), T, 0, stream>>>(t2, 0.0f, N_NODES * HID);
        gnn_edge_agg128<<<nb(N_EDGES * 32, T), T, 0, stream>>>(t1, src, dst, t2, N_EDGES);
        gnn_post_norm_relu<<<nb(N_NODES * HID, T), T, 0, stream>>>(t2, dins, b2, t0);
    }
    // GAT: z = t0 @ Wg ; edge softmax over dst ; oagg ; head mean -> t1
    {
        int waves = (N_NODES / 16) * (ZDIM / 16);         // 625*32 = 20000
        gnn_gemm_wmma_f32<<<nb(waves * 32, T), T, 0, stream>>>(t0, Wg, nullptr, zbuf,
                                                               N_NODES, HID, ZDIM);
        gnn_attn_proj<<<nb(N_NODES * HEADS, T), T, 0, stream>>>(zbuf, al, ar, el, er);
        gnn_fill_f32<<<nb(N_NODES * HEADS, T), T, 0, stream>>>(mmax, -__builtin_huge_valf(),
                                                               N_NODES * HEADS);
        gnn_fill_f32<<<nb(N_NODES * HEADS, T), T, 0, stream>>>(den, 0.0f, N_NODES * HEADS);
        gnn_edge_scores<<<nb(N_EDGES * HEADS, T), T, 0, stream>>>(el, er, src, dst, eedge,
                                                                  mmax, N_EDGES);
        gnn_edge_exp<<<nb(N_EDGES * HEADS, T), T, 0, stream>>>(dst, mmax, eedge, den, N_EDGES);
        gnn_edge_alpha<<<nb(N_EDGES * HEADS, T), T, 0, stream>>>(dst, den, eedge, N_EDGES);
        gnn_fill_f32<<<nb(N_NODES * ZDIM, T), T, 0, stream>>>(oagg, 0.0f, N_NODES * ZDIM);
        gnn_gat_agg<<<nb(N_EDGES * HEADS * 32, T), T, 0, stream>>>(zbuf, eedge, src, dst,
                                                                   oagg, N_EDGES);
        gnn_head_mean<<<nb(N_NODES * HID, T), T, 0, stream>>>(oagg, bg, t1);
    }
    // GraphConv 3 (128 -> 1) + sigmoid
    gnn_conv3_dot<<<nb(N_NODES, T), T, 0, stream>>>(t1, douts, W3, sN);
    gnn_fill_f32<<<nb(N_NODES, T), T, 0, stream>>>(sAgg, 0.0f, N_NODES);
    gnn_edge_agg1<<<nb(N_EDGES, T), T, 0, stream>>>(sN, src, dst, sAgg, N_EDGES);
    gnn_final_sigmoid<<<nb(N_NODES, T), T, 0, stream>>>(sAgg, dins, b3, out);
}